// EpistemicValue_24043226923381
// MI455X (gfx1250) — compile-verified
//
#include <hip/hip_runtime.h>
#include <math.h>

// ---------------------------------------------------------------------------
// EpistemicValue for MI455X (gfx1250, wave32).
// Memory-bound (~360 MB traffic -> ~15us @ 23.3 TB/s). One lane per dim
// element; V_WMMA_F32_16X16X4_F32 (full fp32 accumulate) used for the
// deterministic intra-wave reduction of info_gain.
// ---------------------------------------------------------------------------

typedef __attribute__((ext_vector_type(2))) float v2f;
typedef __attribute__((ext_vector_type(8))) float v8f;

#define NBINS   16
#define TAU     0.05f
#define EPS_SUM 1e-8f
#define EPS_LOG 1e-10f

// Sum of one float per lane across the wave via WMMA f32 16x16x4 with B=ones.
// A layout (32-bit A 16x4): VGPR0 holds K=0 (lanes 0-15) / K=2 (lanes 16-31),
// VGPR1 holds K=1/K=3 (we set it to 0). With B all-ones:
//   D[m][n] = IG_m + IG_{m+16}   (rows replicated across n)
// Lane n<16 holds D rows 0..7 in its 8 D VGPRs, lane n>=16 holds rows 8..15,
// so per-lane sum + shfl_xor(16) gives the full 32-lane total on every lane.
// Requires EXEC == all ones (call from reconverged code only).
__device__ __forceinline__ float wave_sum_wmma(float v) {
  v2f a; a[0] = v;    a[1] = 0.0f;
  v2f b; b[0] = 1.0f; b[1] = 1.0f;   // ones matrix: layout-independent
  v8f c = {};
  c = __builtin_amdgcn_wmma_f32_16x16x4_f32(
      /*neg_a=*/false, a, /*neg_b=*/false, b,
      /*c_mod=*/(short)0, c, /*reuse_a=*/false, /*reuse_b=*/false);
  float s = ((c[0] + c[1]) + (c[2] + c[3])) + ((c[4] + c[5]) + (c[6] + c[7]));
  s += __shfl_xor(s, 16, 32);
  return s;
}

__global__ void __launch_bounds__(256)
epistemic_main(const float* __restrict__ obs,      // (8, dim)
               const float* __restrict__ counts,   // (dim, 16)
               float* __restrict__ out,            // flat tuple output
               float* __restrict__ partials,       // (gridDim.x,) block sums
               int dim) {
  const int d = blockIdx.x * 256 + threadIdx.x;
  const bool active = d < dim;

  float ig = 0.0f;
  if (active) {
    // ---- observation mean over 8 rows (coalesced column loads) ----
    float om = 0.0f;
#pragma unroll
    for (int r = 0; r < 8; ++r) om += obs[(size_t)r * (size_t)dim + d];
    om *= 0.125f;

    // sigmoid binning
    float nrm = 1.0f / (1.0f + __expf(-om));
    int bin = (int)(nrm * (float)(NBINS - 1));
    bin = bin < 0 ? 0 : (bin > NBINS - 1 ? NBINS - 1 : bin);

    // ---- belief row: 64 contiguous bytes per lane, 4x b128 loads ----
    const float4* c4 = (const float4*)(counts + (size_t)d * NBINS);
    float4 v0 = c4[0], v1 = c4[1], v2 = c4[2], v3 = c4[3];
    float c[NBINS] = {v0.x, v0.y, v0.z, v0.w, v1.x, v1.y, v1.z, v1.w,
                      v2.x, v2.y, v2.z, v2.w, v3.x, v3.y, v3.z, v3.w};

    float S = 0.0f;
#pragma unroll
    for (int i = 0; i < NBINS; ++i) S += c[i];
    const float rP = 1.0f / fmaxf(S, EPS_SUM);          // prior normalizer
    const float rQ = 1.0f / fmaxf(S + 1.0f, EPS_SUM);   // posterior normalizer

    float Hp = 0.0f, Hq = 0.0f;
    float nc[NBINS];
#pragma unroll
    for (int i = 0; i < NBINS; ++i) {
      const float oh = (i == bin) ? 1.0f : 0.0f;
      const float p  = c[i] * rP;
      const float q  = (c[i] + oh) * rQ;
      const float lp = __log2f(p + EPS_LOG);
      const float lq = __log2f(q + EPS_LOG);
      Hp -= p * lp;              // prior entropy
      Hq -= q * lq;              // posterior entropy
      ig += q * (lq - lp);       // KL info gain (log of ratio = diff of logs)
      nc[i] = fmaxf(c[i] * (1.0f - TAU) + oh * TAU, 0.01f);  // EMA update
    }
    ig = fmaxf(ig, 0.0f);

    // ---- outputs (tuple concat: ig | mean | Hp | Hq | epi | new_counts) ----
    out[d] = ig;
    out[(size_t)dim + 1 + d]     = Hp;
    out[2 * (size_t)dim + 1 + d] = Hq;
    // new_counts base = 3*dim+2 elements -> byte offset == 8 (mod 16):
    // use 8-byte-aligned float2 stores.
    float2* ncp = (float2*)(out + 3 * (size_t)dim + 2) + (size_t)d * 8;
#pragma unroll
    for (int i = 0; i < 8; ++i) ncp[i] = make_float2(nc[2 * i], nc[2 * i + 1]);
  }

  // ---- deterministic block reduction of info_gain ----
  // EXEC is all-ones here (reconverged); inactive lanes contribute ig = 0.
  const float wsum = wave_sum_wmma(ig);

  __shared__ float lds[8];
  const int wave = threadIdx.x >> 5;
  const int lane = threadIdx.x & 31;
  if (lane == 0) lds[wave] = wsum;
  __syncthreads();
  if (threadIdx.x == 0) {
    float t = 0.0f;
#pragma unroll
    for (int i = 0; i < 8; ++i) t += lds[i];   // fixed order -> deterministic
    partials[blockIdx.x] = t;
  }
}

__global__ void __launch_bounds__(256)
epistemic_finalize(const float* __restrict__ partials, int nPart,
                   float* __restrict__ out, int dim) {
  __shared__ float sm[256];
  float s = 0.0f;
  for (int i = threadIdx.x; i < nPart; i += 256) s += partials[i];  // fixed order
  sm[threadIdx.x] = s;
  __syncthreads();
#pragma unroll
  for (int off = 128; off > 0; off >>= 1) {      // fixed tree -> deterministic
    if (threadIdx.x < off) sm[threadIdx.x] += sm[threadIdx.x + off];
    __syncthreads();
  }
  if (threadIdx.x == 0) {
    const float mean = sm[0] / (float)dim;
    const float epi  = 1.0f / (1.0f + __expf(-(mean * 50.0f - 1.0f)));
    out[dim] = mean;                 // mean_info_gain
    out[3 * (size_t)dim + 1] = epi;  // epistemic_value
  }
}

extern "C" void kernel_launch(void* const* d_in, const int* in_sizes, int n_in,
                              void* d_out, int out_size, void* d_ws, size_t ws_size,
                              hipStream_t stream) {
  const float* obs    = (const float*)d_in[0];  // (8, dim) f32
  const float* counts = (const float*)d_in[1];  // (dim, 16) f32
  float* out = (float*)d_out;

  const int dim = in_sizes[1] / NBINS;          // 2,097,152
  const int nBlocks = (dim + 255) / 256;        // 8192 -> 32 KB of d_ws partials
  float* partials = (float*)d_ws;

  epistemic_main<<<nBlocks, 256, 0, stream>>>(obs, counts, out, partials, dim);
  epistemic_finalize<<<1, 256, 0, stream>>>(partials, nBlocks, out, dim);
}